// NaiveVQC_69836168233501
// MI455X (gfx1250) — compile-verified
//
#include <hip/hip_runtime.h>

// ---------------------------------------------------------------------------
// 14-qubit VQC statevector simulator for MI455X (gfx1250, wave32).
//  * One 256-thread workgroup (8 waves) per batch sample.
//  * Full 16384-amplitude complex64 statevector lives in LDS (128 KB < 320 KB).
//  * Rz*Ry*Rx(*Ry_enc) fused per wire; wires fused in triplets -> 8x8 complex
//    gate -> 16x16 real embedding -> 4 chained v_wmma_f32_16x16x4_f32.
//  * CNOT ring = in-place disjoint LDS swaps (pure permutation, no FLOPs).
// ---------------------------------------------------------------------------

typedef float v2f __attribute__((ext_vector_type(2)));
typedef float v8f __attribute__((ext_vector_type(8)));

#define NWIRES  14
#define NSTATE  (1 << NWIRES)        // 16384 amplitudes
#define NLAYERS 4
#define TPB     256                  // 8 wave32 waves
#define NGROUPS 5                    // 4 wire-triplets + (12,13,identity)
#define NBATCH  (NSTATE / 8 / 16)    // 128 column-batches per 3-wire gate

static __device__ __forceinline__ float2 cmul(float2 a, float2 b) {
  return make_float2(a.x * b.x - a.y * b.y, a.x * b.y + a.y * b.x);
}
static __device__ __forceinline__ float2 cadd(float2 a, float2 b) {
  return make_float2(a.x + b.x, a.y + b.y);
}
// 2x2 complex matmul, row-major float2[4]: C = A*B
static __device__ __forceinline__ void mm2(const float2* A, const float2* B, float2* C) {
  C[0] = cadd(cmul(A[0], B[0]), cmul(A[1], B[2]));
  C[1] = cadd(cmul(A[0], B[1]), cmul(A[1], B[3]));
  C[2] = cadd(cmul(A[2], B[0]), cmul(A[3], B[2]));
  C[3] = cadd(cmul(A[2], B[1]), cmul(A[3], B[3]));
}
// insert a zero bit at (final) position p
static __device__ __forceinline__ unsigned ins0(unsigned x, int p) {
  unsigned m = (1u << p) - 1u;
  return ((x & ~m) << 1) | (x & m);
}
// amplitude offset of sub-index j (3 bits) at bit positions pa>pb>pc-ish
static __device__ __forceinline__ unsigned offj(int j, int pa, int pb, int pc) {
  return (((unsigned)(j >> 2) & 1u) << pa) | (((unsigned)(j >> 1) & 1u) << pb) |
         (((unsigned)j & 1u) << pc);
}
static __device__ __forceinline__ v8f wmma_f32(v2f a, v2f b, v8f c) {
  // (neg_a, A, neg_b, B, c_mod, C, reuse_a, reuse_b)
  return __builtin_amdgcn_wmma_f32_16x16x4_f32(false, a, false, b, (short)0, c,
                                               false, false);
}

__global__ void __launch_bounds__(TPB)
vqc_wmma_kernel(const float* __restrict__ x_raw, const float* __restrict__ angles,
                float* __restrict__ out) {
  extern __shared__ char smem[];
  float2* psi  = (float2*)smem;                                  // 16384 * 8 B
  float*  psif = (float*)smem;                                   // scalar alias
  float*  g8re = (float*)(smem + (size_t)NSTATE * sizeof(float2)); // 5*64 f
  float*  g8im = g8re + NGROUPS * 64;                              // 5*64 f
  float2* w2   = (float2*)(g8im + NGROUPS * 64);                   // 14*4 cplx
  float*  red  = (float*)(w2 + NWIRES * 4);                        // 256 f

  const int t    = threadIdx.x;
  const int samp = blockIdx.x;
  const int lane = t & 31;
  const int wave = t >> 5;     // 0..7
  const int col  = lane & 15;  // matrix column / A-row owned by this lane
  const int half = lane >> 4;  // 0: K/Re-half, 1: K-hi/Im-half

  // |00...0>
  for (int i = t; i < NSTATE; i += TPB)
    psi[i] = make_float2(i == 0 ? 1.0f : 0.0f, 0.0f);
  __syncthreads();

  for (int layer = 0; layer < NLAYERS; ++layer) {
    // ---- fused per-wire 2x2 gate: G = Rz*Ry*Rx*(layer==0 ? Ry_enc : I) ----
    if (t < NWIRES) {
      const int w = t;
      float ax = angles[((layer * NWIRES) + w) * 3 + 0];
      float ay = angles[((layer * NWIRES) + w) * 3 + 1];
      float az = angles[((layer * NWIRES) + w) * 3 + 2];
      float cx = cosf(0.5f * ax), sx = sinf(0.5f * ax);
      float2 RX[4] = {{cx, 0.f}, {0.f, -sx}, {0.f, -sx}, {cx, 0.f}};
      float cy = cosf(0.5f * ay), sy = sinf(0.5f * ay);
      float2 RY[4] = {{cy, 0.f}, {-sy, 0.f}, {sy, 0.f}, {cy, 0.f}};
      float cz = cosf(0.5f * az), sz = sinf(0.5f * az);
      float2 RZ[4] = {{cz, -sz}, {0.f, 0.f}, {0.f, 0.f}, {cz, sz}};
      float2 M0[4], M1[4], G[4];
      if (layer == 0) {
        float xe = tanhf(x_raw[samp * NWIRES + w]) * 3.14159265358979f;
        float ce = cosf(0.5f * xe), se = sinf(0.5f * xe);
        float2 RE[4] = {{ce, 0.f}, {-se, 0.f}, {se, 0.f}, {ce, 0.f}};
        mm2(RX, RE, M0);
        mm2(RY, M0, M1);
        mm2(RZ, M1, G);
      } else {
        mm2(RY, RX, M1);
        mm2(RZ, M1, G);
      }
      w2[w * 4 + 0] = G[0]; w2[w * 4 + 1] = G[1];
      w2[w * 4 + 2] = G[2]; w2[w * 4 + 3] = G[3];
    }
    __syncthreads();

    // ---- 8x8 complex gate per group (group 4: I (x) G12 (x) G13) ----
    for (int e = t; e < NGROUPS * 64; e += TPB) {
      int g = e >> 6, idx = e & 63;
      int i = idx >> 3, j = idx & 7;
      int i2 = (i >> 2) & 1, i1 = (i >> 1) & 1, i0 = i & 1;
      int j2 = (j >> 2) & 1, j1 = (j >> 1) & 1, j0 = j & 1;
      float2 v;
      if (g < 4) {
        float2 a = w2[(3 * g + 0) * 4 + i2 * 2 + j2];
        float2 b = w2[(3 * g + 1) * 4 + i1 * 2 + j1];
        float2 c = w2[(3 * g + 2) * 4 + i0 * 2 + j0];
        v = cmul(cmul(a, b), c);
      } else if (i2 == j2) {
        float2 b = w2[12 * 4 + i1 * 2 + j1];
        float2 c = w2[13 * 4 + i0 * 2 + j0];
        v = cmul(b, c);
      } else {
        v = make_float2(0.f, 0.f);
      }
      g8re[e] = v.x;
      g8im[e] = v.y;
    }
    __syncthreads();

    // ---- apply group gates: D(16x16) = G16 x P via 4x v_wmma_f32_16x16x4 ----
#pragma unroll
    for (int g = 0; g < NGROUPS; ++g) {
      // wire w <-> bit (13-w); group 4 pads identity on bit 13
      const int pa = (g < 4) ? (13 - 3 * g) : 13;
      const int pb = (g < 4) ? (12 - 3 * g) : 1;
      const int pc = (g < 4) ? (11 - 3 * g) : 0;

      // A fragments: real embedding [[Re,-Im],[Im,Re]] of 8x8 complex gate.
      // A 16x4 f32 layout: row m = lane%16; V0 holds K=2*half, V1 holds K=2*half+1.
      v2f A[4];
#pragma unroll
      for (int sI = 0; sI < 4; ++sI) {
#pragma unroll
        for (int cI = 0; cI < 2; ++cI) {
          int k = 4 * sI + 2 * half + cI;
          int m = col;
          float re = g8re[g * 64 + (m & 7) * 8 + (k & 7)];
          float im = g8im[g * 64 + (m & 7) * 8 + (k & 7)];
          A[sI][cI] = (m < 8) ? ((k < 8) ? re : -im) : ((k < 8) ? im : re);
        }
      }

      // Amplitudes this lane feeds into B (B 4x16: col=lane%16, K-pair by half)
      const unsigned o_a = offj(2 * half + 0, pa, pb, pc);
      const unsigned o_b = offj(2 * half + 1, pa, pb, pc);
      const unsigned o_c = offj(4 + 2 * half + 0, pa, pb, pc);
      const unsigned o_d = offj(4 + 2 * half + 1, pa, pb, pc);

      for (int bt = wave; bt < NBATCH; bt += TPB / 32) {
        unsigned base = (unsigned)(bt * 16 + col);
        base = ins0(base, pc);
        base = ins0(base, pb);
        base = ins0(base, pa);

        float2 p0 = psi[base + o_a];
        float2 p1 = psi[base + o_b];
        float2 p2 = psi[base + o_c];
        float2 p3 = psi[base + o_d];
        v2f B0 = {p0.x, p1.x};  // K 0..3   : Re rows
        v2f B1 = {p2.x, p3.x};  // K 4..7   : Re rows
        v2f B2 = {p0.y, p1.y};  // K 8..11  : Im rows
        v2f B3 = {p2.y, p3.y};  // K 12..15 : Im rows

        v8f d = {0.f, 0.f, 0.f, 0.f, 0.f, 0.f, 0.f, 0.f};
        d = wmma_f32(A[0], B0, d);
        d = wmma_f32(A[1], B1, d);
        d = wmma_f32(A[2], B2, d);
        d = wmma_f32(A[3], B3, d);

        // D layout: VGPR r = row r (+8 for lanes 16-31), col = lane%16.
        // half==0 lanes hold new Re of o(0..7); half==1 lanes hold new Im.
#pragma unroll
        for (int r = 0; r < 8; ++r) {
          unsigned amp = base + offj(r, pa, pb, pc);
          psif[2u * amp + (unsigned)half] = d[r];
        }
      }
      __syncthreads();
    }

    // ---- CNOT ring: in-place disjoint swaps where control=1, target flips ----
    for (int i = 0; i < NWIRES; ++i) {
      const int cw = i, tw = (i + 1) % NWIRES;
      const int pcb = 13 - cw, ptb = 13 - tw;
      const int hi = pcb > ptb ? pcb : ptb;
      const int lo = pcb ^ ptb ^ hi;
      const unsigned cmask = 1u << pcb, tmask = 1u << ptb;
      for (int k = t; k < (NSTATE >> 2); k += TPB) {
        unsigned x = ins0(ins0((unsigned)k, lo), hi) | cmask;  // ctrl=1, tgt=0
        float2 u = psi[x];
        float2 v = psi[x | tmask];
        psi[x] = v;
        psi[x | tmask] = u;
      }
      __syncthreads();
    }
  }

  // ---- <Z_0>: wire 0 is MSB (bit 13); +|a|^2 if bit=0 else -|a|^2 ----
  float acc = 0.f;
  for (int i = t; i < NSTATE; i += TPB) {
    float2 a = psi[i];
    float p = a.x * a.x + a.y * a.y;
    acc += (i & (1 << 13)) ? -p : p;
  }
  red[t] = acc;
  __syncthreads();
  for (int stride = TPB / 2; stride > 0; stride >>= 1) {
    if (t < stride) red[t] += red[t + stride];
    __syncthreads();
  }
  if (t == 0) out[samp] = red[0];
}

extern "C" void kernel_launch(void* const* d_in, const int* in_sizes, int n_in,
                              void* d_out, int out_size, void* d_ws, size_t ws_size,
                              hipStream_t stream) {
  (void)n_in; (void)d_ws; (void)ws_size; (void)out_size;
  const float* x_raw  = (const float*)d_in[0];   // [bsz, 14] f32
  const float* angles = (const float*)d_in[1];   // [4, 14, 3] f32
  float* out = (float*)d_out;                    // [bsz] f32
  const int bsz = in_sizes[0] / NWIRES;          // 1024

  const size_t smem = (size_t)NSTATE * sizeof(float2)  // statevector   131072 B
                    + (size_t)NGROUPS * 64 * 4 * 2     // g8 re/im        2560 B
                    + (size_t)NWIRES * 4 * 8           // wire gates       448 B
                    + (size_t)TPB * 4;                 // reduction       1024 B

  // >64KB dynamic LDS opt-in (deterministic, capture-safe host call)
  hipFuncSetAttribute((const void*)vqc_wmma_kernel,
                      hipFuncAttributeMaxDynamicSharedMemorySize, (int)smem);

  vqc_wmma_kernel<<<bsz, TPB, smem, stream>>>(x_raw, angles, out);
}